// MambaLM_73478300500304
// MI455X (gfx1250) — compile-verified
//
#include <hip/hip_runtime.h>
#include <hip/hip_bf16.h>
#include <math.h>

#define D_MODEL 768
#define D_STATE 16
#define D_CONV  4
#define N_LAYERS 2
#define VOCAB   32000
#define D_INNER 1536
#define DT_RANK 48
#define SEQ     2048
#define DBC_W   (DT_RANK + 2 * D_STATE)   // 80

typedef float v2f __attribute__((ext_vector_type(2)));
typedef float v8f __attribute__((ext_vector_type(8)));

__device__ __forceinline__ float silu_f(float x) {
    return x / (1.0f + expf(-x));
}
__device__ __forceinline__ float softplus_f(float x) {
    // jax.nn.softplus: max(x,0) + log1p(exp(-|x|))
    return fmaxf(x, 0.0f) + log1pf(expf(-fabsf(x)));
}

// ---------------------------------------------------------------- embedding
__global__ void k_embed(const int* __restrict__ ids, const float* __restrict__ emb,
                        float* __restrict__ x) {
    int l = blockIdx.x;
    int id = ids[l];
    const float* src = emb + (size_t)id * D_MODEL;
    float* dst = x + (size_t)l * D_MODEL;
    for (int i = threadIdx.x; i < D_MODEL; i += blockDim.x) dst[i] = src[i];
}

// ---------------------------------------------------------------- layernorm
__global__ void k_layernorm(const float* __restrict__ x, const float* __restrict__ w,
                            const float* __restrict__ b, float* __restrict__ out) {
    __shared__ float sh_s[256];
    __shared__ float sh_q[256];
    int row = blockIdx.x, tid = threadIdx.x;
    const float* xr = x + (size_t)row * D_MODEL;
    float s = 0.f, q = 0.f;
    for (int i = tid; i < D_MODEL; i += 256) { float v = xr[i]; s += v; q += v * v; }
    sh_s[tid] = s; sh_q[tid] = q;
    __syncthreads();
    for (int off = 128; off > 0; off >>= 1) {
        if (tid < off) { sh_s[tid] += sh_s[tid + off]; sh_q[tid] += sh_q[tid + off]; }
        __syncthreads();
    }
    float mu   = sh_s[0] * (1.0f / D_MODEL);
    float var  = sh_q[0] * (1.0f / D_MODEL) - mu * mu;
    float rstd = rsqrtf(var + 1e-5f);
    float* orow = out + (size_t)row * D_MODEL;
    for (int i = tid; i < D_MODEL; i += 256)
        orow[i] = (xr[i] - mu) * rstd * w[i] + b[i];
}

// ---------------------------------------------------------------- WMMA GEMM
// C[M,N] = A[M,K] * W[N,K]^T   (both K-contiguous, "NT" layout)
// Wave computes a 16x(16*NSUB) strip; one A fragment reused across NSUB B tiles.
// NSUB is compile-time: no predication in the hot loop, EXEC stays all-ones.
// The only guard is a whole-wave scalar early-exit (wave id via readfirstlane
// so the compiler proves uniformity and emits s_cbranch, not exec masking).
// EPI: 0 = store, 1 = softplus(x + bias[n]) store, 2 = C += x (residual).
template <int EPI, int NSUB>
__global__ void k_wmma_gemm(const float* __restrict__ A, const float* __restrict__ W,
                            const float* __restrict__ bias, float* __restrict__ C,
                            int N, int K, int lda, int ldw, int ldc) {
    const int lane = threadIdx.x & 31;
    const int wave = __builtin_amdgcn_readfirstlane(threadIdx.x) >> 5;  // uniform
    const int tm   = blockIdx.x;
    const int tn0  = (blockIdx.y * 4 + wave) * NSUB;
    if ((tn0 + NSUB) * 16 > N) return;               // scalar branch, whole wave

    const int nl = lane & 15;
    const int kh = (lane >> 4) << 1;                 // 0 or 2
    const float* ap  = A + (size_t)(tm * 16 + nl) * lda + kh;
    const float* wp0 = W + (size_t)(tn0 * 16 + nl) * ldw + kh;
    const size_t wstride = (size_t)16 * ldw;

    v8f acc[NSUB];
#pragma unroll
    for (int j = 0; j < NSUB; ++j)
        acc[j] = (v8f){0.f, 0.f, 0.f, 0.f, 0.f, 0.f, 0.f, 0.f};

    for (int k = 0; k < K; k += 4) {
        v2f a = *(const v2f*)(ap + k);
#pragma unroll
        for (int j = 0; j < NSUB; ++j) {
            v2f bm = *(const v2f*)(wp0 + (size_t)j * wstride + k);
            acc[j] = __builtin_amdgcn_wmma_f32_16x16x4_f32(
                false, a, false, bm, (short)0, acc[j], false, false);
        }
    }

    // C/D layout: VGPR v, lane l -> (m = v + 8*(l>>4), n = l&15)
    const int mbase = tm * 16 + ((lane >> 4) << 3);
#pragma unroll
    for (int j = 0; j < NSUB; ++j) {
        int ncol = (tn0 + j) * 16 + nl;
        float bv = (EPI == 1) ? bias[ncol] : 0.0f;
#pragma unroll
        for (int v = 0; v < 8; ++v) {
            size_t idx = (size_t)(mbase + v) * ldc + ncol;
            float val = acc[j][v];
            if (EPI == 1)      C[idx] = softplus_f(val + bv);
            else if (EPI == 2) C[idx] += val;
            else               C[idx] = val;
        }
    }
}

// ---------------------------------------------------------------- depthwise causal conv + silu
__global__ void k_conv_silu(const float* __restrict__ xz, const float* __restrict__ cw,
                            const float* __restrict__ cb, float* __restrict__ out) {
    int d = blockIdx.x * blockDim.x + threadIdx.x;
    int l = blockIdx.y;
    float val = cb[d];
    const float* w = cw + d * D_CONV;
#pragma unroll
    for (int k = 0; k < D_CONV; ++k) {
        int ll = l + k - (D_CONV - 1);
        if (ll >= 0) val += xz[(size_t)ll * (2 * D_INNER) + d] * w[k];
    }
    out[(size_t)l * D_INNER + d] = silu_f(val);
}

// ---------------------------------------------------------------- selective scan (fused D-skip + silu(z) gate)
__global__ void k_scan(const float* __restrict__ xc, const float* __restrict__ dt,
                       const float* __restrict__ dbc, const float* __restrict__ xz,
                       const float* __restrict__ A_log, const float* __restrict__ Dsk,
                       float* __restrict__ y) {
    int d = blockIdx.x * blockDim.x + threadIdx.x;
    float A[D_STATE], h[D_STATE];
#pragma unroll
    for (int n = 0; n < D_STATE; ++n) {
        A[n] = -expf(A_log[d * D_STATE + n]);
        h[n] = 0.0f;
    }
    float dval = Dsk[d];
    for (int l = 0; l < SEQ; ++l) {
        float xv  = xc[(size_t)l * D_INNER + d];
        float dtv = dt[(size_t)l * D_INNER + d];
        float dtx = dtv * xv;
        float acc = xv * dval;
        const float* Bp = dbc + (size_t)l * DBC_W + DT_RANK;
        const float* Cp = Bp + D_STATE;
#pragma unroll
        for (int n = 0; n < D_STATE; ++n) {
            h[n] = expf(dtv * A[n]) * h[n] + dtx * Bp[n];
            acc += Cp[n] * h[n];
        }
        float zv = xz[(size_t)l * (2 * D_INNER) + D_INNER + d];
        y[(size_t)l * D_INNER + d] = acc * silu_f(zv);
    }
}

// ---------------------------------------------------------------- driver
extern "C" void kernel_launch(void* const* d_in, const int* in_sizes, int n_in,
                              void* d_out, int out_size, void* d_ws, size_t ws_size,
                              hipStream_t stream) {
    (void)in_sizes; (void)n_in; (void)out_size; (void)ws_size;
    const int*   ids   = (const int*)d_in[0];
    const float* emb   = (const float*)d_in[1];
    const float* nw    = (const float*)d_in[2];
    const float* nb    = (const float*)d_in[3];
    const float* inw   = (const float*)d_in[4];
    const float* cw    = (const float*)d_in[5];
    const float* cb    = (const float*)d_in[6];
    const float* xpw   = (const float*)d_in[7];
    const float* dtw   = (const float*)d_in[8];
    const float* dtb   = (const float*)d_in[9];
    const float* alog  = (const float*)d_in[10];
    const float* dskip = (const float*)d_in[11];
    const float* outw  = (const float*)d_in[12];
    const float* lnfw  = (const float*)d_in[13];
    const float* lnfb  = (const float*)d_in[14];
    float* logits = (float*)d_out;

    float* ws    = (float*)d_ws;
    float* xres  = ws;  ws += (size_t)SEQ * D_MODEL;      // residual stream
    float* lnb   = ws;  ws += (size_t)SEQ * D_MODEL;      // layernorm output
    float* xzb   = ws;  ws += (size_t)SEQ * 2 * D_INNER;  // in_proj output (x_ssm | z)
    float* xcv   = ws;  ws += (size_t)SEQ * D_INNER;      // conv+silu output
    float* dbc   = ws;  ws += (size_t)SEQ * DBC_W;        // x_proj output (dt|B|C)
    float* dtbuf = ws;  ws += (size_t)SEQ * D_INNER;      // softplus(dt)
    float* ybuf  = ws;  ws += (size_t)SEQ * D_INNER;      // scan output

    dim3 blk(128);
    k_embed<<<SEQ, 256, 0, stream>>>(ids, emb, xres);

    for (int L = 0; L < N_LAYERS; ++L) {
        k_layernorm<<<SEQ, 256, 0, stream>>>(xres, nw + L * D_MODEL, nb + L * D_MODEL, lnb);

        { // in_proj: [2048,3072] = lnb[2048,768] * inw[3072,768]^T ; Ntiles=192=12*16
            int N = 2 * D_INNER, K = D_MODEL;
            dim3 g(SEQ / 16, 192 / 16);
            k_wmma_gemm<0, 4><<<g, blk, 0, stream>>>(lnb, inw + (size_t)L * N * K, nullptr,
                                                     xzb, N, K, K, K, N);
        }
        k_conv_silu<<<dim3(D_INNER / 256, SEQ), 256, 0, stream>>>(
            xzb, cw + (size_t)L * D_INNER * D_CONV, cb + L * D_INNER, xcv);

        { // x_proj: [2048,80] = xcv[2048,1536] * xpw[80,1536]^T ; Ntiles=5 -> NSUB=1
            int N = DBC_W, K = D_INNER;
            dim3 g(SEQ / 16, 2);   // 8 wave-slots, 5 used (waves 5..7 exit via scalar branch)
            k_wmma_gemm<0, 1><<<g, blk, 0, stream>>>(xcv, xpw + (size_t)L * N * K, nullptr,
                                                     dbc, N, K, K, K, N);
        }
        { // dt_proj + bias + softplus: [2048,1536]; A row stride 80, K=48 ; Ntiles=96=6*16
            int N = D_INNER, K = DT_RANK;
            dim3 g(SEQ / 16, 96 / 16);
            k_wmma_gemm<1, 4><<<g, blk, 0, stream>>>(dbc, dtw + (size_t)L * N * K,
                                                     dtb + L * D_INNER, dtbuf,
                                                     N, K, DBC_W, K, N);
        }
        k_scan<<<D_INNER / 256, 256, 0, stream>>>(
            xcv, dtbuf, dbc, xzb,
            alog + (size_t)L * D_INNER * D_STATE, dskip + L * D_INNER, ybuf);

        { // out_proj + residual add: xres += ybuf[2048,1536] * outw[768,1536]^T ; Ntiles=48=3*16
            int N = D_MODEL, K = D_INNER;
            dim3 g(SEQ / 16, 48 / 16);
            k_wmma_gemm<2, 4><<<g, blk, 0, stream>>>(ybuf, outw + (size_t)L * N * K, nullptr,
                                                     xres, N, K, K, K, N);
        }
    }

    k_layernorm<<<SEQ, 256, 0, stream>>>(xres, lnfw, lnfb, lnb);

    { // tied LM head: logits[2048,32000] = lnb[2048,768] * emb[32000,768]^T ; Ntiles=2000=125*16
        int N = VOCAB, K = D_MODEL;
        dim3 g(SEQ / 16, 2000 / 16);
        k_wmma_gemm<0, 4><<<g, blk, 0, stream>>>(lnb, emb, nullptr, logits, N, K, K, K, N);
    }
}